// BayesianAtlas_74277164417758
// MI455X (gfx1250) — compile-verified
//
#include <hip/hip_runtime.h>

#define THREADS 256

// 8-byte pair with 4-byte alignment so the backend may emit an unaligned
// global_load_b64 (gfx1250 supports unaligned dword-pair loads) or split b32s.
typedef struct __attribute__((aligned(4))) { float x, y; } f2u;

#if defined(__HIP_DEVICE_COMPILE__)
typedef __attribute__((address_space(1))) int* gi32p;   // global int*
typedef __attribute__((address_space(3))) int* li32p;   // LDS int*

__device__ __forceinline__ void wait_async0() {
#if __has_builtin(__builtin_amdgcn_s_wait_asynccnt)
    __builtin_amdgcn_s_wait_asynccnt(0);
#else
    asm volatile("s_wait_asynccnt 0" ::: "memory");
#endif
}
#endif

__global__ __launch_bounds__(THREADS)
void trilerp3d_kernel(const float* __restrict__ vel,   // [B,3,G,G,G]
                      const float* __restrict__ pts,   // [B,N,3]
                      const float* __restrict__ bb,    // [3,2]
                      const int*   __restrict__ gsz,   // scalar G
                      float* __restrict__ out,         // [B,N,3]
                      unsigned long long velTotal,
                      unsigned long long ptsTotal)
{
    __shared__ float lds[THREADS * 3];

    const int G = gsz[0];
    const unsigned long long G3 = (unsigned long long)G * (unsigned)G * (unsigned)G;
    const unsigned int B = (unsigned int)(velTotal / (3ull * G3));
    const unsigned int N = (unsigned int)(ptsTotal / (3ull * (unsigned long long)B));
    const unsigned int totalPts = B * N;
    const unsigned int totalDw  = totalPts * 3u;

    const unsigned int tid       = threadIdx.x;
    const unsigned int blockBase = blockIdx.x * THREADS;  // first point of block
    const unsigned int dwBase    = blockBase * 3u;        // first flat dword of block

#if defined(__HIP_DEVICE_COMPILE__)
    // ---- Stage this block's points into LDS: coalesced async global->LDS ----
#if __has_builtin(__builtin_amdgcn_global_load_async_to_lds_b32)
#pragma unroll
    for (int k = 0; k < 3; ++k) {
        unsigned int d = tid + (unsigned)k * THREADS;
        if (dwBase + d < totalDw) {
            __builtin_amdgcn_global_load_async_to_lds_b32(
                (gi32p)((void*)(pts + dwBase + d)),
                (li32p)((void*)&lds[d]),
                0, 0);
        }
    }
    wait_async0();
#else
#pragma unroll
    for (int k = 0; k < 3; ++k) {
        unsigned int d = tid + (unsigned)k * THREADS;
        if (dwBase + d < totalDw) lds[d] = pts[dwBase + d];
    }
#endif
#else
    for (int k = 0; k < 3; ++k) {
        unsigned int d = tid + (unsigned)k * THREADS;
        if (dwBase + d < totalDw) lds[d] = pts[dwBase + d];
    }
#endif
    __syncthreads();

    // ---- Trilinear interpolation ----
    const unsigned int pid = blockBase + tid;
    float a0 = 0.0f, a1 = 0.0f, a2 = 0.0f;
    if (pid < totalPts) {
        const unsigned int b = pid / N;

        const float px = lds[tid * 3 + 0];
        const float py = lds[tid * 3 + 1];
        const float pz = lds[tid * 3 + 2];

        const float Gm1f = (float)(G - 1);
        const float lo0 = bb[0], hi0 = bb[1];
        const float lo1 = bb[2], hi1 = bb[3];
        const float lo2 = bb[4], hi2 = bb[5];

        float u = (px - lo0) / (hi0 - lo0) * Gm1f;
        float v = (py - lo1) / (hi1 - lo1) * Gm1f;
        float w = (pz - lo2) / (hi2 - lo2) * Gm1f;

        float fu1 = fminf(fmaxf(floorf(u), 0.0f), Gm1f);
        float fv1 = fminf(fmaxf(floorf(v), 0.0f), Gm1f);
        float fw1 = fminf(fmaxf(floorf(w), 0.0f), Gm1f);

        float fu = u - fu1, gu = fu1 + 1.0f - u;
        float fv = v - fv1, gv = fv1 + 1.0f - v;
        float fw = w - fw1, gw = fw1 + 1.0f - w;

        int iu1 = (int)fu1, iv1 = (int)fv1, iw1 = (int)fw1;
        int iu2 = (iu1 + 1 > G - 1) ? (G - 1) : iu1 + 1;
        int iv2 = (iv1 + 1 > G - 1) ? (G - 1) : iv1 + 1;
        int iw2 = (iw1 + 1 > G - 1) ? (G - 1) : iw1 + 1;

        // Pair the two w-adjacent corners into one 8-byte load.
        // At the top boundary (iw2==iw1==G-1) shift the base down by one and
        // move all weight onto the high element: exact, no OOB access.
        int   sh  = (iw2 == iw1) ? 1 : 0;
        int   wb  = iw1 - sh;
        float wlo = sh ? 0.0f : gw;
        float whi = sh ? (gw + fw) : fw;

        const unsigned int o11 = ((unsigned)(iu1 * G + iv1)) * (unsigned)G + (unsigned)wb;
        const unsigned int o12 = ((unsigned)(iu1 * G + iv2)) * (unsigned)G + (unsigned)wb;
        const unsigned int o21 = ((unsigned)(iu2 * G + iv1)) * (unsigned)G + (unsigned)wb;
        const unsigned int o22 = ((unsigned)(iu2 * G + iv2)) * (unsigned)G + (unsigned)wb;

        const float c11 = gu * gv, c12 = gu * fv, c21 = fu * gv, c22 = fu * fv;

        const float* velB = vel + (unsigned long long)b * 3ull * G3;
        float acc[3];
#pragma unroll
        for (int c = 0; c < 3; ++c) {
            const float* base = velB + (unsigned long long)c * G3;
            f2u q11 = *(const f2u*)(base + o11);
            f2u q12 = *(const f2u*)(base + o12);
            f2u q21 = *(const f2u*)(base + o21);
            f2u q22 = *(const f2u*)(base + o22);
            acc[c] = c11 * (q11.x * wlo + q11.y * whi)
                   + c12 * (q12.x * wlo + q12.y * whi)
                   + c21 * (q21.x * wlo + q21.y * whi)
                   + c22 * (q22.x * wlo + q22.y * whi);
        }
        a0 = acc[0]; a1 = acc[1]; a2 = acc[2];
    }

    // ---- Transpose results through LDS for coalesced output stores ----
    // Each thread writes only its own 3 slots (read only by itself above),
    // so no barrier is needed before these writes.
    lds[tid * 3 + 0] = a0;
    lds[tid * 3 + 1] = a1;
    lds[tid * 3 + 2] = a2;
    __syncthreads();

#if defined(__HIP_DEVICE_COMPILE__)
#if __has_builtin(__builtin_amdgcn_global_store_async_from_lds_b32)
#pragma unroll
    for (int k = 0; k < 3; ++k) {
        unsigned int d = tid + (unsigned)k * THREADS;
        if (dwBase + d < totalDw) {
            __builtin_amdgcn_global_store_async_from_lds_b32(
                (gi32p)((void*)(out + dwBase + d)),
                (li32p)((void*)&lds[d]),
                0, 0);
        }
    }
    wait_async0();   // (S_ENDPGM also does an implicit wait-idle)
#else
#pragma unroll
    for (int k = 0; k < 3; ++k) {
        unsigned int d = tid + (unsigned)k * THREADS;
        if (dwBase + d < totalDw)
            __builtin_nontemporal_store(lds[d], out + dwBase + d);  // keep L2 for the field
    }
#endif
#else
    for (int k = 0; k < 3; ++k) {
        unsigned int d = tid + (unsigned)k * THREADS;
        if (dwBase + d < totalDw) out[dwBase + d] = lds[d];
    }
#endif
}

extern "C" void kernel_launch(void* const* d_in, const int* in_sizes, int n_in,
                              void* d_out, int out_size, void* d_ws, size_t ws_size,
                              hipStream_t stream) {
    (void)n_in; (void)out_size; (void)d_ws; (void)ws_size;
    const float* vel = (const float*)d_in[0];
    const float* pts = (const float*)d_in[1];
    const float* bb  = (const float*)d_in[2];
    const int*   gsz = (const int*)d_in[3];
    float* out = (float*)d_out;

    const unsigned long long velTotal = (unsigned long long)in_sizes[0];
    const unsigned long long ptsTotal = (unsigned long long)in_sizes[1];
    const unsigned long long totalPts = ptsTotal / 3ull;
    const unsigned int blocks = (unsigned int)((totalPts + THREADS - 1) / THREADS);

    hipLaunchKernelGGL(trilerp3d_kernel, dim3(blocks), dim3(THREADS), 0, stream,
                       vel, pts, bb, gsz, out, velTotal, ptsTotal);
}